// GCNLayer_4037269258345
// MI455X (gfx1250) — compile-verified
//
#include <hip/hip_runtime.h>
#include <stdint.h>

// GCN layer: h = feature @ W + b ; out = segment_sum(h[src], dst)
// Stage 1: zero d_out
// Stage 2: fp32 WMMA GEMM (16x16x4), 32 rows/wave, bias folded into acc init, h -> d_ws
//          - feature streamed with non-temporal loads (preserve L2 for scatter phase)
//          - W staged into LDS via GLOBAL_LOAD_ASYNC_TO_LDS_B32 (ASYNCcnt path)
// Stage 3: edge scatter with scalar index loads + global f32 atomics (L2-resident working set)

typedef __attribute__((ext_vector_type(2))) float v2f;
typedef __attribute__((ext_vector_type(8))) float v8f;

#define D_IN   256
#define D_OUT  64
#define KCHUNK 64
#define NPAIRS (KCHUNK / 2)   // 32 K-pairs per chunk
#define PSTR   160            // floats per pair-row: 128 data + 32 pad; 160 % 64 == 32 -> half-waves on disjoint banks

// ---------------- zero d_out ----------------
__global__ void gcn_zero_kernel(float4* __restrict__ p, int n4) {
    int i = blockIdx.x * blockDim.x + threadIdx.x;
    if (i < n4) p[i] = make_float4(0.f, 0.f, 0.f, 0.f);
}

// ---------------- GEMM + bias: h = feature @ W + b ----------------
// One wave computes a 32x64 tile: two 16-row A fragments share four B fragments.
// LDS holds W K-chunk in pair-interleaved form: lw[p*PSTR + 2c + parity] = W[2p+parity][c]
// so each B fragment {W[brow][c], W[brow+1][c]} is a single aligned ds_load_b64.
__global__ void __launch_bounds__(160)
gcn_gemm_bias_kernel(const float* __restrict__ A,     // [M, 256]
                     const float* __restrict__ W,     // [256, 64]
                     const float* __restrict__ bias,  // [64]
                     float* __restrict__ H,           // [M, 64]
                     int ntiles32) {
    __shared__ float lw[NPAIRS * PSTR];   // 20 KB

    const int tid   = threadIdx.x;
    const int wave  = tid >> 5;
    const int lane  = tid & 31;
    const int m16   = lane & 15;   // row (A) / col (B,C)
    const int khalf = lane >> 4;   // K-pair selector within 4-wide slab

    const int  tile     = blockIdx.x * 5 + wave;
    const bool active   = (tile < ntiles32);
    const int  row_base = active ? tile * 32 : 0;   // clamp so inactive waves load valid addrs

    const float* arow0 = A + (size_t)(row_base + m16) * D_IN + khalf * 2;
    const float* arow1 = arow0 + (size_t)16 * D_IN;

    // LDS byte offset of lw[] for the async-to-LDS path
    uint32_t lw_base;
    {
        __attribute__((address_space(3))) float* p3 =
            (__attribute__((address_space(3))) float*)lw;
        lw_base = (uint32_t)(uintptr_t)p3;
    }

    // bias folded into accumulator init: h = A@W + b
    const float bv0 = bias[m16];
    const float bv1 = bias[16 + m16];
    const float bv2 = bias[32 + m16];
    const float bv3 = bias[48 + m16];

    v8f acc00, acc01, acc02, acc03, acc10, acc11, acc12, acc13;
    #pragma unroll
    for (int r = 0; r < 8; ++r) {
        acc00[r] = bv0; acc01[r] = bv1; acc02[r] = bv2; acc03[r] = bv3;
        acc10[r] = bv0; acc11[r] = bv1; acc12[r] = bv2; acc13[r] = bv3;
    }

    for (int k0 = 0; k0 < D_IN; k0 += KCHUNK) {
        __syncthreads();   // protect lw reads of previous chunk
        // stage W[k0 .. k0+63][0..63] into pair-interleaved LDS via async DMA
        // (uniform per-wave trip counts; ASYNCcnt-tracked, no VGPR round-trip)
        for (int i = tid; i < KCHUNK * D_OUT; i += 160) {
            int r = i >> 6;
            int c = i & 63;
            uint32_t lds_addr = lw_base + 4u * (uint32_t)((r >> 1) * PSTR + 2 * c + (r & 1));
            const float* gsrc = W + (size_t)(k0 + r) * D_OUT + c;
            asm volatile("global_load_async_to_lds_b32 %0, %1, off"
                         :: "v"(lds_addr), "v"(gsrc)
                         : "memory");
        }
        asm volatile("s_wait_asynccnt 0x0" ::: "memory");
        __syncthreads();

        #pragma unroll 4
        for (int kk = 0; kk < KCHUNK; kk += 4) {
            // A 16x4 fragments: VGPR0 = K(khalf*2), VGPR1 = K(khalf*2+1)
            // Non-temporal: feature is read exactly once; don't evict h/out from L2.
            const v2f a0 = __builtin_nontemporal_load((const v2f*)(arow0 + k0 + kk));
            const v2f a1 = __builtin_nontemporal_load((const v2f*)(arow1 + k0 + kk));

            // B 4x16 fragments: pair p = kk/2 + khalf -> rows {kk+2*khalf, +1}, one b64 each
            const float* lb = &lw[((kk >> 1) + khalf) * PSTR + 2 * m16];
            const v2f b0 = *(const v2f*)(lb + 0);
            const v2f b1 = *(const v2f*)(lb + 32);
            const v2f b2 = *(const v2f*)(lb + 64);
            const v2f b3 = *(const v2f*)(lb + 96);

            acc00 = __builtin_amdgcn_wmma_f32_16x16x4_f32(false, a0, false, b0, (short)0, acc00, false, false);
            acc01 = __builtin_amdgcn_wmma_f32_16x16x4_f32(false, a0, false, b1, (short)0, acc01, false, false);
            acc02 = __builtin_amdgcn_wmma_f32_16x16x4_f32(false, a0, false, b2, (short)0, acc02, false, false);
            acc03 = __builtin_amdgcn_wmma_f32_16x16x4_f32(false, a0, false, b3, (short)0, acc03, false, false);
            acc10 = __builtin_amdgcn_wmma_f32_16x16x4_f32(false, a1, false, b0, (short)0, acc10, false, false);
            acc11 = __builtin_amdgcn_wmma_f32_16x16x4_f32(false, a1, false, b1, (short)0, acc11, false, false);
            acc12 = __builtin_amdgcn_wmma_f32_16x16x4_f32(false, a1, false, b2, (short)0, acc12, false, false);
            acc13 = __builtin_amdgcn_wmma_f32_16x16x4_f32(false, a1, false, b3, (short)0, acc13, false, false);
        }
    }

    if (!active) return;

    // C/D layout: VGPR r -> row (r + khalf*8), col m16 (+ 16*j per accumulator)
    float* hp0 = H + (size_t)(row_base + khalf * 8) * D_OUT + m16;
    float* hp1 = hp0 + (size_t)16 * D_OUT;
    #pragma unroll
    for (int r = 0; r < 8; ++r) {
        hp0[(size_t)r * D_OUT + 0]  = acc00[r];
        hp0[(size_t)r * D_OUT + 16] = acc01[r];
        hp0[(size_t)r * D_OUT + 32] = acc02[r];
        hp0[(size_t)r * D_OUT + 48] = acc03[r];
        hp1[(size_t)r * D_OUT + 0]  = acc10[r];
        hp1[(size_t)r * D_OUT + 16] = acc11[r];
        hp1[(size_t)r * D_OUT + 32] = acc12[r];
        hp1[(size_t)r * D_OUT + 48] = acc13[r];
    }
}

// ---------------- scatter: out[dst[e]] += h[src[e]] ----------------
__global__ void __launch_bounds__(256)
gcn_scatter_kernel(const float* __restrict__ h,
                   const int* __restrict__ src,
                   const int* __restrict__ dst,
                   float* __restrict__ out,
                   int nedges) {
    const int wid = blockIdx.x * (blockDim.x >> 5) + (threadIdx.x >> 5);
    const int e   = __builtin_amdgcn_readfirstlane(wid);   // wave-uniform edge id
    if (e >= nedges) return;

    const int lane = threadIdx.x & 31;                     // 2 columns per lane
    const int s = __builtin_amdgcn_readfirstlane(src[e]);  // scalar index loads
    const int d = __builtin_amdgcn_readfirstlane(dst[e]);

    const v2f v = *(const v2f*)&h[(size_t)s * D_OUT + lane * 2];
    float* o = &out[(size_t)d * D_OUT + lane * 2];
    atomicAdd(&o[0], v.x);
    atomicAdd(&o[1], v.y);
}

extern "C" void kernel_launch(void* const* d_in, const int* in_sizes, int n_in,
                              void* d_out, int out_size, void* d_ws, size_t ws_size,
                              hipStream_t stream) {
    const float* feature = (const float*)d_in[0];
    const float* W       = (const float*)d_in[1];
    const float* bias    = (const float*)d_in[2];
    const int*   src     = (const int*)d_in[3];
    const int*   dst     = (const int*)d_in[4];

    const int n_nodes = in_sizes[0] / D_IN;    // 100000
    const int n_edges = in_sizes[3];           // 1600000

    float* out = (float*)d_out;
    float* h   = (float*)d_ws;                 // needs n_nodes * 64 * 4 = 25.6 MB

    // 1) zero the output accumulator
    const int n4 = (n_nodes * D_OUT) / 4;
    gcn_zero_kernel<<<(n4 + 255) / 256, 256, 0, stream>>>((float4*)out, n4);

    // 2) h = feature @ W + b   (one wave per 32-row tile, 5 waves per block)
    const int ntiles32 = n_nodes / 32;         // 3125
    gcn_gemm_bias_kernel<<<(ntiles32 + 4) / 5, 160, 0, stream>>>(feature, W, bias, h, ntiles32);

    // 3) scatter-add over edges (one wave per edge, 8 edges per block)
    gcn_scatter_kernel<<<(n_edges + 7) / 8, 256, 0, stream>>>(h, src, dst, out, n_edges);
}